// CoefficientDotProductAttention_34892314313425
// MI455X (gfx1250) — compile-verified
//
#include <hip/hip_runtime.h>
#include <hip/hip_bf16.h>
#include <math.h>

typedef __attribute__((ext_vector_type(2))) float v2f;
typedef __attribute__((ext_vector_type(8))) float v8f;

#define B_   4
#define N_   2048
#define H_   8
#define BH_  32
#define QT_  128   // queries per workgroup (8 waves x 16)
#define KT_  128   // keys staged in LDS per tile
#define SCALE_ 0.35355339059327373f   // 1/sqrt(8)

// ---------------------------------------------------------------------------
// Kernel 1: grade-wise QKV projection.
// x[b][n][0..7] -> q/k/v laid out as [bh = b*8+h][n][0..7] (f32).
// Grades: dim0 (d=1), dims1-3 (d=3), dims4-6 (d=3), dim7 (d=1).
// W_g has shape [H*d, d] row-major; out[h*d+r] = b[h*d+r] + sum_j x[s+j]*W[(h*d+r)*d+j]
// ---------------------------------------------------------------------------
struct ProjParams {
  const float* W[12];    // Wq0..3, Wk0..3, Wv0..3
  const float* bias[12]; // bq0..3, bk0..3, bv0..3
};

__device__ __forceinline__ void proj8(const float xv[8], int h,
    const float* W0, const float* W1, const float* W2, const float* W3,
    const float* c0, const float* c1, const float* c2, const float* c3,
    float* __restrict__ out) {
  float4 o0, o1;
  o0.x = fmaf(xv[0], W0[h], c0[h]);
  {
    const float* w = W1 + (h*3 + 0)*3;
    o0.y = c1[h*3+0] + xv[1]*w[0] + xv[2]*w[1] + xv[3]*w[2];
    w = W1 + (h*3 + 1)*3;
    o0.z = c1[h*3+1] + xv[1]*w[0] + xv[2]*w[1] + xv[3]*w[2];
    w = W1 + (h*3 + 2)*3;
    o0.w = c1[h*3+2] + xv[1]*w[0] + xv[2]*w[1] + xv[3]*w[2];
  }
  {
    const float* w = W2 + (h*3 + 0)*3;
    o1.x = c2[h*3+0] + xv[4]*w[0] + xv[5]*w[1] + xv[6]*w[2];
    w = W2 + (h*3 + 1)*3;
    o1.y = c2[h*3+1] + xv[4]*w[0] + xv[5]*w[1] + xv[6]*w[2];
    w = W2 + (h*3 + 2)*3;
    o1.z = c2[h*3+2] + xv[4]*w[0] + xv[5]*w[1] + xv[6]*w[2];
  }
  o1.w = fmaf(xv[7], W3[h], c3[h]);
  ((float4*)out)[0] = o0;
  ((float4*)out)[1] = o1;
}

__global__ __launch_bounds__(256)
void qkv_proj_kernel(const float* __restrict__ x, ProjParams P,
                     float* __restrict__ Qg, float* __restrict__ Kg,
                     float* __restrict__ Vg) {
  const int n  = blockIdx.x * blockDim.x + threadIdx.x; // 0..N-1
  const int bh = blockIdx.y;                            // 0..31
  const int b = bh >> 3, h = bh & 7;
  const float4* xr = (const float4*)(x + ((size_t)b * N_ + n) * 8);
  float4 x0 = xr[0], x1 = xr[1];
  float xv[8] = {x0.x, x0.y, x0.z, x0.w, x1.x, x1.y, x1.z, x1.w};
  const size_t off = ((size_t)bh * N_ + n) * 8;
  proj8(xv, h, P.W[0], P.W[1], P.W[2], P.W[3],
               P.bias[0], P.bias[1], P.bias[2], P.bias[3], Qg + off);
  proj8(xv, h, P.W[4], P.W[5], P.W[6], P.W[7],
               P.bias[4], P.bias[5], P.bias[6], P.bias[7], Kg + off);
  proj8(xv, h, P.W[8], P.W[9], P.W[10], P.W[11],
               P.bias[8], P.bias[9], P.bias[10], P.bias[11], Vg + off);
}

// ---------------------------------------------------------------------------
// Kernel 2: flash attention per (bh, 128-query tile).
// Each wave owns 16 queries. S^T(16 keys x 16 queries) = K_tile(16x8) * Q^T(8x16)
// via two exact V_WMMA_F32_16X16X4_F32 (K-dims 0..3 and 4..7).
// C-layout: lane L -> query (L&15); rows r=0..7 -> keys kb*16 + r + 8*(L>>4).
// Online softmax per lane; P*V accumulated per-lane over this half's 8 keys,
// halves merged with shfl_xor(16) at the end.
// ---------------------------------------------------------------------------
__global__ __launch_bounds__(256)
void flash_attn_kernel(const float* __restrict__ Qg,
                       const float* __restrict__ Kg,
                       const float* __restrict__ Vg,
                       const unsigned char* __restrict__ maskb,
                       float* __restrict__ HO) {
  __shared__ float4 Kl4[KT_ * 2];
  __shared__ float4 Vl4[KT_ * 2];
  __shared__ float  Ml[KT_];
  const float* Kl = (const float*)Kl4;

  const int t    = threadIdx.x;
  const int wave = t >> 5;
  const int lane = t & 31;
  const int lq   = lane & 15;        // query-in-tile for this lane
  const int hi   = lane >> 4;        // lane half
  const int c    = hi * 2;           // K-dim pair supplied by this half

  const int nqt = N_ / QT_;
  const int bh  = blockIdx.x / nqt;
  const int qt  = blockIdx.x % nqt;
  const int b   = bh >> 3;

  const int qg = qt * QT_ + wave * 16 + lq;
  const float* qrow = Qg + ((size_t)bh * N_ + qg) * 8;
  v2f bq1, bq2;                      // B fragments (Q^T), reused every tile
  bq1.x = qrow[c];     bq1.y = qrow[c + 1];
  bq2.x = qrow[c + 4]; bq2.y = qrow[c + 5];

  float m = -__builtin_inff();
  float l = 0.f;
  float acc[8];
#pragma unroll
  for (int i = 0; i < 8; ++i) acc[i] = 0.f;

  const float4* Kgr = (const float4*)(Kg + ((size_t)bh * N_) * 8);
  const float4* Vgr = (const float4*)(Vg + ((size_t)bh * N_) * 8);
  const unsigned char* mrow = maskb + (size_t)b * N_;

  for (int kt = 0; kt < N_ / KT_; ++kt) {
    __syncthreads();
    Kl4[t] = Kgr[kt * (KT_ * 2) + t];
    Vl4[t] = Vgr[kt * (KT_ * 2) + t];
    if (t < KT_) Ml[t] = mrow[kt * KT_ + t] ? -__builtin_inff() : 0.f;
    __syncthreads();

#pragma unroll
    for (int kb = 0; kb < KT_ / 16; ++kb) {
      // A fragments: K rows (M = key index within sub-block)
      const float* kp = Kl + (kb * 16 + lq) * 8;
      v2f a1, a2;
      a1.x = kp[c];     a1.y = kp[c + 1];
      a2.x = kp[c + 4]; a2.y = kp[c + 5];
      v8f s = {};
      s = __builtin_amdgcn_wmma_f32_16x16x4_f32(false, a1, false, bq1,
                                                (short)0, s, false, false);
      s = __builtin_amdgcn_wmma_f32_16x16x4_f32(false, a2, false, bq2,
                                                (short)0, s, false, false);

      const int kbase = kb * 16 + hi * 8;   // first key row held by this lane
      float sv[8];
#pragma unroll
      for (int r = 0; r < 8; ++r) sv[r] = s[r] * SCALE_ + Ml[kbase + r];

      float bm = sv[0];
#pragma unroll
      for (int r = 1; r < 8; ++r) bm = fmaxf(bm, sv[r]);
      bm = fmaxf(bm, __shfl_xor(bm, 16, 32));   // merge key halves (same query)
      const float mnew  = fmaxf(m, bm);
      const float alpha = __expf(m - mnew);     // exp(-inf)=0 on first block
      float p[8], ps = 0.f;
#pragma unroll
      for (int r = 0; r < 8; ++r) { p[r] = __expf(sv[r] - mnew); ps += p[r]; }
      l = l * alpha + ps;                       // per-half row-sum
#pragma unroll
      for (int i = 0; i < 8; ++i) acc[i] *= alpha;
#pragma unroll
      for (int r = 0; r < 8; ++r) {
        float4 v0 = Vl4[(kbase + r) * 2 + 0];   // LDS broadcast within half
        float4 v1 = Vl4[(kbase + r) * 2 + 1];
        acc[0] = fmaf(p[r], v0.x, acc[0]);
        acc[1] = fmaf(p[r], v0.y, acc[1]);
        acc[2] = fmaf(p[r], v0.z, acc[2]);
        acc[3] = fmaf(p[r], v0.w, acc[3]);
        acc[4] = fmaf(p[r], v1.x, acc[4]);
        acc[5] = fmaf(p[r], v1.y, acc[5]);
        acc[6] = fmaf(p[r], v1.z, acc[6]);
        acc[7] = fmaf(p[r], v1.w, acc[7]);
      }
      m = mnew;
    }
  }

  // Merge the two key-halves (lane L <-> L^16 hold the same query).
  const float lt = l + __shfl_xor(l, 16, 32);
#pragma unroll
  for (int i = 0; i < 8; ++i) acc[i] += __shfl_xor(acc[i], 16, 32);
  if (lane < 16) {
    const float inv = 1.f / lt;
    float4 o0 = {acc[0]*inv, acc[1]*inv, acc[2]*inv, acc[3]*inv};
    float4 o1 = {acc[4]*inv, acc[5]*inv, acc[6]*inv, acc[7]*inv};
    float4* orow = (float4*)(HO + ((size_t)bh * N_ + qg) * 8);
    orow[0] = o0;
    orow[1] = o1;
  }
}

// ---------------------------------------------------------------------------
// Kernel 3: grade-wise output projection.
// out[b][n][s+r] = bo_g[r] + sum_h sum_j HO[b*8+h][n][s+j] * Wo_g[r][h*d + j]
// ---------------------------------------------------------------------------
struct OutParams {
  const float* Wo[4];
  const float* bo[4];
};

__global__ __launch_bounds__(256)
void out_proj_kernel(const float* __restrict__ HO, OutParams P,
                     float* __restrict__ out) {
  const int idx = blockIdx.x * blockDim.x + threadIdx.x; // 0..B*N-1
  if (idx >= B_ * N_) return;
  const int b = idx / N_, n = idx % N_;
  float a0 = P.bo[0][0];
  float a1 = P.bo[1][0], a2 = P.bo[1][1], a3 = P.bo[1][2];
  float a4 = P.bo[2][0], a5 = P.bo[2][1], a6 = P.bo[2][2];
  float a7 = P.bo[3][0];
  const float* W1 = P.Wo[1]; // [3][24]
  const float* W2 = P.Wo[2]; // [3][24]
#pragma unroll
  for (int h = 0; h < H_; ++h) {
    const float4* ho = (const float4*)(HO + (((size_t)(b*H_ + h)) * N_ + n) * 8);
    float4 h0 = ho[0], h1 = ho[1];
    a0 = fmaf(h0.x, P.Wo[0][h], a0);
    a1 += h0.y*W1[0*24 + h*3+0] + h0.z*W1[0*24 + h*3+1] + h0.w*W1[0*24 + h*3+2];
    a2 += h0.y*W1[1*24 + h*3+0] + h0.z*W1[1*24 + h*3+1] + h0.w*W1[1*24 + h*3+2];
    a3 += h0.y*W1[2*24 + h*3+0] + h0.z*W1[2*24 + h*3+1] + h0.w*W1[2*24 + h*3+2];
    a4 += h1.x*W2[0*24 + h*3+0] + h1.y*W2[0*24 + h*3+1] + h1.z*W2[0*24 + h*3+2];
    a5 += h1.x*W2[1*24 + h*3+0] + h1.y*W2[1*24 + h*3+1] + h1.z*W2[1*24 + h*3+2];
    a6 += h1.x*W2[2*24 + h*3+0] + h1.y*W2[2*24 + h*3+1] + h1.z*W2[2*24 + h*3+2];
    a7 = fmaf(h1.w, P.Wo[3][h], a7);
  }
  float4* orow = (float4*)(out + ((size_t)b * N_ + n) * 8);
  orow[0] = make_float4(a0, a1, a2, a3);
  orow[1] = make_float4(a4, a5, a6, a7);
}

// ---------------------------------------------------------------------------
// Input order (setup_inputs() insertion order):
// 0: x [4,2048,8] f32       1: key_padding_mask [4,2048] bool
// 2..5:  Wq grades   6..9:  bq grades
// 10..13:Wk          14..17:bk
// 18..21:Wv          22..25:bv
// 26..29:Wo          30..33:bo
// Workspace: Q,K,V,HO each BH*N*8 f32 = 2 MB -> 8 MB total.
// ---------------------------------------------------------------------------
extern "C" void kernel_launch(void* const* d_in, const int* in_sizes, int n_in,
                              void* d_out, int out_size, void* d_ws, size_t ws_size,
                              hipStream_t stream) {
  (void)in_sizes; (void)n_in; (void)out_size; (void)ws_size;
  const float* x = (const float*)d_in[0];
  const unsigned char* mask = (const unsigned char*)d_in[1];

  ProjParams P;
  OutParams  OP;
  for (int g = 0; g < 4; ++g) {
    P.W[g]        = (const float*)d_in[2 + g];   // Wq
    P.bias[g]     = (const float*)d_in[6 + g];   // bq
    P.W[4 + g]    = (const float*)d_in[10 + g];  // Wk
    P.bias[4 + g] = (const float*)d_in[14 + g];  // bk
    P.W[8 + g]    = (const float*)d_in[18 + g];  // Wv
    P.bias[8 + g] = (const float*)d_in[22 + g];  // bv
    OP.Wo[g]      = (const float*)d_in[26 + g];
    OP.bo[g]      = (const float*)d_in[30 + g];
  }

  float* ws = (float*)d_ws;
  const size_t arr = (size_t)BH_ * N_ * 8;
  float* Qg = ws;
  float* Kg = ws + arr;
  float* Vg = ws + 2 * arr;
  float* HO = ws + 3 * arr;

  dim3 g1(N_ / 256, BH_);
  qkv_proj_kernel<<<g1, 256, 0, stream>>>(x, P, Qg, Kg, Vg);
  flash_attn_kernel<<<BH_ * (N_ / QT_), 256, 0, stream>>>(Qg, Kg, Vg, mask, HO);
  out_proj_kernel<<<(B_ * N_ + 255) / 256, 256, 0, stream>>>(HO, OP, (float*)d_out);
}